// RoIHeads_82575041232910
// MI455X (gfx1250) — compile-verified
//
#include <hip/hip_runtime.h>
#include <stdint.h>

#define N_ROI 1024
#define K_KP  17
#define HM    56
#define ROW   3136              // 56*56
#define ROWS  17408             // 1024*17
#define WPB   8                 // waves per block
#define KP_BLOCKS (ROWS / WPB)  // 2176
#define LOG2E 1.4426950408889634f
#define LN2   0.6931471805599453f
#define BETA  (1.0f/9.0f)

#define AS1 __attribute__((address_space(1)))
#define AS3 __attribute__((address_space(3)))

typedef int v4i __attribute__((vector_size(16)));
typedef int v2i __attribute__((vector_size(8)));

#if defined(__AMDGCN__) && \
    __has_builtin(__builtin_amdgcn_global_load_async_to_lds_b128) && \
    __has_builtin(__builtin_amdgcn_global_load_async_to_lds_b64)
#define KP_ASYNC 1
#endif

__device__ __forceinline__ float wexp2(float x) { return __builtin_amdgcn_exp2f(x); }
__device__ __forceinline__ float wlog2(float x) { return __builtin_amdgcn_logf(x); }

template <int N>
__device__ __forceinline__ void wait_async_le() {
#ifdef KP_ASYNC
#if __has_builtin(__builtin_amdgcn_s_wait_asynccnt)
  __builtin_amdgcn_s_wait_asynccnt(N);
#else
  asm volatile("s_wait_asynccnt %0" :: "i"(N) : "memory");
#endif
#endif
}

// Online softmax update in log2 domain: y = x*log2(e)
__device__ __forceinline__ void osm_update(float xv, float& m, float& s) {
  float y = xv * LOG2E;
  if (y > m) { s *= wexp2(m - y); m = y; }   // rare rescale: ~log(n) times/lane
  s += wexp2(y - m);
}

// ---------------- keypoint CE: one wave per row of 3136 ----------------
// Dominant kernel: streams 214 MB of keypoint_logits once (HBM roofline ~9.2us)
// through LDS via double-buffered async copies (ASYNCcnt pipeline).
__global__ void __launch_bounds__(256)
kp_loss_kernel(const float* __restrict__ kpl, const float* __restrict__ proposals,
               const float* __restrict__ gtkp, const int* __restrict__ matched,
               float* __restrict__ partials) {
  __shared__ __align__(16) float buf[WPB][2][256];  // double buffer, 1KB/chunk/wave
  __shared__ float rn[WPB], rc[WPB];

  const int wid = threadIdx.x >> 5;
  const int lane = threadIdx.x & 31;
  const int r = blockIdx.x * WPB + wid;  // row in [0, 17408)
  const float* rowp = kpl + (size_t)r * ROW;
  float* row_nc = const_cast<float*>(rowp);

  float m = -1.0e30f, s = 0.0f;

#ifdef KP_ASYNC
  // issue = two b128s (256 floats) per chunk; 12 full chunks + 64-float b64 tail
#define KP_ISSUE2(c)                                                          \
  do {                                                                        \
    __builtin_amdgcn_global_load_async_to_lds_b128(                           \
        (AS1 v4i*)(row_nc + (c) * 256 + lane * 4),                            \
        (AS3 v4i*)&buf[wid][(c) & 1][lane * 4], 0, 0);                        \
    __builtin_amdgcn_global_load_async_to_lds_b128(                           \
        (AS1 v4i*)(row_nc + (c) * 256 + 128 + lane * 4),                      \
        (AS3 v4i*)&buf[wid][(c) & 1][128 + lane * 4], 0, 0);                  \
  } while (0)
  KP_ISSUE2(0);  // kick off the DMA pipeline immediately
  for (int j = 0; j < 11; ++j) {
    KP_ISSUE2(j + 1);
    wait_async_le<2>();  // in-order completion => chunk j (2 loads) has landed
    const float4 va = *(const float4*)&buf[wid][j & 1][lane * 4];
    const float4 vb = *(const float4*)&buf[wid][j & 1][128 + lane * 4];
    const float vv[8] = {va.x, va.y, va.z, va.w, vb.x, vb.y, vb.z, vb.w};
#pragma unroll
    for (int c = 0; c < 8; ++c) osm_update(vv[c], m, s);
  }
  // epilogue: chunk 11 + tail
  __builtin_amdgcn_global_load_async_to_lds_b64(
      (AS1 v2i*)(row_nc + 3072 + lane * 2),
      (AS3 v2i*)&buf[wid][0][lane * 2], 0, 0);
  wait_async_le<1>();  // chunk 11 done (tail may be in flight)
  {
    const float4 va = *(const float4*)&buf[wid][1][lane * 4];
    const float4 vb = *(const float4*)&buf[wid][1][128 + lane * 4];
    const float vv[8] = {va.x, va.y, va.z, va.w, vb.x, vb.y, vb.z, vb.w};
#pragma unroll
    for (int c = 0; c < 8; ++c) osm_update(vv[c], m, s);
  }
  wait_async_le<0>();  // tail done
  {
    const float2 v = *(const float2*)&buf[wid][0][lane * 2];
    osm_update(v.x, m, s);
    osm_update(v.y, m, s);
  }
#undef KP_ISSUE2
#else
  for (int base = lane * 4; base < ROW; base += 128) {  // coalesced b128 fallback
    const float4 v = *(const float4*)(rowp + base);
    const float vv[4] = {v.x, v.y, v.z, v.w};
#pragma unroll
    for (int c = 0; c < 4; ++c) osm_update(vv[c], m, s);
  }
#endif

  // ---- target computation (after the stream; loads overlap wave drain) ----
  const int n = r / K_KP;
  const int k = r - n * K_KP;
  const int g = matched[n];
  const float kx  = gtkp[(g * K_KP + k) * 3 + 0];
  const float ky  = gtkp[(g * K_KP + k) * 3 + 1];
  const float vis = gtkp[(g * K_KP + k) * 3 + 2];
  const float p0 = proposals[n * 4 + 0], p1 = proposals[n * 4 + 1];
  const float p2 = proposals[n * 4 + 2], p3 = proposals[n * 4 + 3];
  const float sx = (float)HM / (p2 - p0);
  const float sy = (float)HM / (p3 - p1);
  int xi = (int)floorf((kx - p0) * sx);
  int yi = (int)floorf((ky - p1) * sy);
  if (kx == p2) xi = HM - 1;
  if (ky == p3) yi = HM - 1;
  const bool valid = (xi >= 0) && (yi >= 0) && (xi < HM) && (yi < HM) && (vis > 0.0f);
  const int tgt = valid ? (yi * HM + xi) : 0;
  const float xt = rowp[tgt];   // broadcast load; row is L2-resident

  // ---- merge (m, s) across the wave (wave32 butterfly) ----
#pragma unroll
  for (int off = 16; off > 0; off >>= 1) {
    float m2 = __shfl_xor(m, off, 32);
    float s2 = __shfl_xor(s, off, 32);
    float mm = fmaxf(m, m2);
    s = s * wexp2(m - mm) + s2 * wexp2(m2 - mm);
    m = mm;
  }
  const float nll = LN2 * (m + wlog2(s)) - xt;

  if (lane == 0) { rn[wid] = valid ? nll : 0.0f; rc[wid] = valid ? 1.0f : 0.0f; }
  __syncthreads();
  if (threadIdx.x == 0) {
    float A = 0, B = 0;
    for (int w = 0; w < WPB; ++w) { A += rn[w]; B += rc[w]; }
    partials[blockIdx.x * 2 + 0] = A;
    partials[blockIdx.x * 2 + 1] = B;
  }
}

// ---------------- classification + box losses (tiny) ----------------
__global__ void __launch_bounds__(1024)
cls_box_kernel(const float* __restrict__ logits, const float* __restrict__ boxreg,
               const float* __restrict__ regtgt, const int* __restrict__ labels,
               float* __restrict__ wsf) {
  __shared__ float red[32][4];
  const int n = threadIdx.x;  // one thread per ROI, N=1024
  const float4 a = *(const float4*)(logits + n * 8);
  const float4 b = *(const float4*)(logits + n * 8 + 4);
  float x[8] = {a.x, a.y, a.z, a.w, b.x, b.y, b.z, b.w};
  const int L = labels[n];

  float mx = x[0];
#pragma unroll
  for (int c = 1; c < 8; ++c) mx = fmaxf(mx, x[c]);
  float s = 0.0f;
#pragma unroll
  for (int c = 0; c < 8; ++c) s += wexp2((x[c] - mx) * LOG2E);
  const float sel = logits[n * 8 + L];
  const float nll = mx + LN2 * wlog2(s) - sel;
  const bool keep = (L != 7);  // IGNORE_LABEL
  float cn = keep ? nll : 0.0f;
  float cc = keep ? 1.0f : 0.0f;

  float bsum = 0.0f;
  const float* brp = boxreg + n * 32 + L * 4;
  const float* rtp = regtgt + n * 4;
#pragma unroll
  for (int j = 0; j < 4; ++j) {
    float d = brp[j] - rtp[j];
    float ad = fabsf(d);
    bsum += (ad < BETA) ? (4.5f * d * d) : (ad - 0.5f * BETA);
  }
  float bl = (L > 0) ? bsum : 0.0f;

#pragma unroll
  for (int off = 16; off > 0; off >>= 1) {   // wave32 reduction
    cn += __shfl_xor(cn, off, 32);
    cc += __shfl_xor(cc, off, 32);
    bl += __shfl_xor(bl, off, 32);
  }
  const int wid = threadIdx.x >> 5, lane = threadIdx.x & 31;
  if (lane == 0) { red[wid][0] = cn; red[wid][1] = cc; red[wid][2] = bl; }
  __syncthreads();
  if (threadIdx.x == 0) {
    float A = 0, B = 0, Cc = 0;
    for (int w = 0; w < 32; ++w) { A += red[w][0]; B += red[w][1]; Cc += red[w][2]; }
    wsf[0] = A; wsf[1] = B; wsf[2] = Cc;
  }
}

// ---------------- deterministic final reduction ----------------
__global__ void __launch_bounds__(256)
finalize_kernel(const float* __restrict__ wsf, const float* __restrict__ partials,
                float* __restrict__ out) {
  __shared__ float sn[256], sc[256];
  const int t = threadIdx.x;
  float a = 0.0f, b = 0.0f;
  for (int i = t; i < KP_BLOCKS; i += 256) { a += partials[2 * i]; b += partials[2 * i + 1]; }
  sn[t] = a; sc[t] = b;
  __syncthreads();
  for (int st = 128; st >= 1; st >>= 1) {
    if (t < st) { sn[t] += sn[t + st]; sc[t] += sc[t + st]; }
    __syncthreads();
  }
  if (t == 0) {
    out[0] = wsf[0] / fmaxf(wsf[1], 1.0f);     // classification loss
    out[1] = wsf[2] / (float)N_ROI;            // box loss
    out[2] = sn[0] / fmaxf(sc[0], 1.0f);       // keypoint loss
  }
}

extern "C" void kernel_launch(void* const* d_in, const int* in_sizes, int n_in,
                              void* d_out, int out_size, void* d_ws, size_t ws_size,
                              hipStream_t stream) {
  const float* class_logits       = (const float*)d_in[0];
  const float* box_regression     = (const float*)d_in[1];
  const float* regression_targets = (const float*)d_in[2];
  const float* proposals          = (const float*)d_in[3];
  const float* gt_keypoints       = (const float*)d_in[4];
  const float* keypoint_logits    = (const float*)d_in[5];
  const int*   labels             = (const int*)d_in[6];
  const int*   matched            = (const int*)d_in[7];

  float* wsf      = (float*)d_ws;   // [0..2]: cls_sum, cls_cnt, box_sum
  float* partials = wsf + 16;       // 2176 * 2 floats of per-block kp partials
  float* out      = (float*)d_out;  // 3 floats

  cls_box_kernel<<<1, 1024, 0, stream>>>(class_logits, box_regression,
                                         regression_targets, labels, wsf);
  kp_loss_kernel<<<KP_BLOCKS, 256, 0, stream>>>(keypoint_logits, proposals,
                                                gt_keypoints, matched, partials);
  finalize_kernel<<<1, 256, 0, stream>>>(wsf, partials, out);
}